// GNN1_56667798503738
// MI455X (gfx1250) — compile-verified
//
#include <hip/hip_runtime.h>

// ---------------------------------------------------------------------------
// GNN forward (GraphConv x2 + GCNConv + Linear head) for MI455X / gfx1250.
// fp32 end-to-end; dense GEMMs via V_WMMA_F32_16X16X4_F32 (wave32 tensor op),
// weights staged block-wide into LDS via GLOBAL_LOAD_ASYNC_TO_LDS_B128,
// graph aggregation via native global_atomic_add_f32.
// ---------------------------------------------------------------------------

#define NN 100000
#define EE 800000
#define FDIM 64
#define CDIM 10
#define BLK 256

typedef float v2f __attribute__((ext_vector_type(2)));
typedef float v8f __attribute__((ext_vector_type(8)));

// ------------------------- LDS staging helpers -----------------------------

#if defined(__gfx1250__) && __has_builtin(__builtin_amdgcn_global_load_async_to_lds_b128)
#define USE_ASYNC_LDS 1
typedef int async_v4i __attribute__((vector_size(16)));
#else
#define USE_ASYNC_LDS 0
#endif

__device__ __forceinline__ void stage_to_lds(float* lds, const float* g,
                                             int nfloats, int t) {
#if USE_ASYNC_LDS
    auto gp = (__attribute__((address_space(1))) async_v4i*)g;
    auto lp = (__attribute__((address_space(3))) async_v4i*)lds;
    for (int i = t; i < (nfloats >> 2); i += BLK)
        __builtin_amdgcn_global_load_async_to_lds_b128(gp + i, lp + i, 0, 0);
#else
    const float4* g4 = (const float4*)g;
    float4* l4 = (float4*)lds;
    for (int i = t; i < (nfloats >> 2); i += BLK) l4[i] = g4[i];
#endif
}

__device__ __forceinline__ void stage_wait() {
#if USE_ASYNC_LDS
#if __has_builtin(__builtin_amdgcn_s_wait_asynccnt)
    __builtin_amdgcn_s_wait_asynccnt(0);
#else
    asm volatile("s_wait_asynccnt 0" ::: "memory");
#endif
#endif
    __syncthreads();
}

// ----------------------------- utility kernels -----------------------------

__global__ void zero_f4_kernel(float4* __restrict__ p, int n4) {
    int i = blockIdx.x * blockDim.x + threadIdx.x;
    if (i < n4) p[i] = make_float4(0.f, 0.f, 0.f, 0.f);
}

__global__ void fill1_kernel(float* __restrict__ p, int n) {
    int i = blockIdx.x * blockDim.x + threadIdx.x;
    if (i < n) p[i] = 1.0f;
}

__global__ void rsqrt_kernel(float* __restrict__ p, int n) {
    int i = blockIdx.x * blockDim.x + threadIdx.x;
    if (i < n) p[i] = rsqrtf(p[i]);
}

// ------------------------------- scatter ops -------------------------------

// One wave (32 lanes) per edge; each lane handles 2 features (64 total).
__global__ void scatter_feat_kernel(const float* __restrict__ feat,
                                    const long long* __restrict__ eidx,
                                    float* __restrict__ acc, int nedges) {
    int tid = blockIdx.x * blockDim.x + threadIdx.x;
    int edge = tid >> 5;
    if (edge >= nedges) return;
    int lane = tid & 31;
    int s = (int)eidx[edge];
    int d = (int)eidx[nedges + edge];
    int f = lane * 2;
    const float2 v = *(const float2*)(feat + (size_t)s * FDIM + f);
    float* dp = acc + (size_t)d * FDIM + f;
    unsafeAtomicAdd(dp, v.x);
    unsafeAtomicAdd(dp + 1, v.y);
}

// Degree accumulation: one thread per edge.
__global__ void scatter_deg_kernel(const long long* __restrict__ eidx,
                                   float* __restrict__ deg, int nedges) {
    int e = blockIdx.x * blockDim.x + threadIdx.x;
    if (e >= nedges) return;
    int d = (int)eidx[nedges + e];
    unsafeAtomicAdd(deg + d, 1.0f);
}

// GCN normalized scatter: acc[dst] += xw[src] * dinv[src] * dinv[dst]
__global__ void scatter_norm_kernel(const float* __restrict__ xw,
                                    const long long* __restrict__ eidx,
                                    const float* __restrict__ dinv,
                                    float* __restrict__ acc, int nedges) {
    int tid = blockIdx.x * blockDim.x + threadIdx.x;
    int edge = tid >> 5;
    if (edge >= nedges) return;
    int lane = tid & 31;
    int s = (int)eidx[edge];
    int d = (int)eidx[nedges + edge];
    float nrm = dinv[s] * dinv[d];
    int f = lane * 2;
    const float2 v = *(const float2*)(xw + (size_t)s * FDIM + f);
    float* dp = acc + (size_t)d * FDIM + f;
    unsafeAtomicAdd(dp, v.x * nrm);
    unsafeAtomicAdd(dp + 1, v.y * nrm);
}

// --------------------------- WMMA fp32 GEMM path ---------------------------
//
// V_WMMA_F32_16X16X4_F32: A = 16x4 f32 (2 VGPRs), B = 4x16 f32 (2 VGPRs),
// C/D = 16x16 f32 (8 VGPRs). ISA layouts:
//   A: lanes 0-15 hold M=lane, K={0,1}; lanes 16-31 hold M=lane-16, K={2,3}
//   B: lanes 0-15 hold N=lane rows K={0,1}; lanes 16-31 N=lane-16, K={2,3}
//   C: VGPR r, lanes 0-15 -> (M=r, N=lane); lanes 16-31 -> (M=8+r, N=lane-16)
//
// One wave computes a 16x64 output tile: 4 accumulators, 16 K-steps.
// Weights live in LDS, shared by all 8 waves of the block.

// out = act( A1 @ W1 + A2 @ W2 + bias ), all [*,64]x[64,64]
__global__ void gemm_dual_kernel(const float* __restrict__ A1,
                                 const float* __restrict__ W1,
                                 const float* __restrict__ A2,
                                 const float* __restrict__ W2,
                                 const float* __restrict__ bias,
                                 float* __restrict__ out, int nrows, int relu) {
    __shared__ float lw1[FDIM * FDIM];
    __shared__ float lw2[FDIM * FDIM];
    __shared__ float lb[FDIM];
    int t = threadIdx.x;
    stage_to_lds(lw1, W1, FDIM * FDIM, t);
    stage_to_lds(lw2, W2, FDIM * FDIM, t);
    stage_to_lds(lb, bias, FDIM, t);
    stage_wait();

    int wave = (blockIdx.x * blockDim.x + t) >> 5;
    int m0 = wave * 16;
    if (m0 + 16 > nrows) return;  // after barrier: safe
    int lane = t & 31;
    int half = lane >> 4;
    int lm = lane & 15;

    v8f acc[4] = {};
    const float* a1row = A1 + (size_t)(m0 + lm) * FDIM;
    const float* a2row = A2 + (size_t)(m0 + lm) * FDIM;

    for (int kk = 0; kk < FDIM; kk += 4) {
        int ko = kk + half * 2;
        v2f a1 = *(const v2f*)(a1row + ko);
        v2f a2 = *(const v2f*)(a2row + ko);
#pragma unroll
        for (int nt = 0; nt < 4; ++nt) {
            int n = nt * 16 + lm;
            v2f b1;
            b1.x = lw1[ko * FDIM + n];
            b1.y = lw1[(ko + 1) * FDIM + n];
            v2f b2;
            b2.x = lw2[ko * FDIM + n];
            b2.y = lw2[(ko + 1) * FDIM + n];
            acc[nt] = __builtin_amdgcn_wmma_f32_16x16x4_f32(
                false, a1, false, b1, (short)0, acc[nt], false, false);
            acc[nt] = __builtin_amdgcn_wmma_f32_16x16x4_f32(
                false, a2, false, b2, (short)0, acc[nt], false, false);
        }
    }

#pragma unroll
    for (int nt = 0; nt < 4; ++nt) {
        int col = nt * 16 + lm;
        float bb = lb[col];
#pragma unroll
        for (int r = 0; r < 8; ++r) {
            int row = m0 + r + half * 8;
            float v = acc[nt][r] + bb;
            if (relu) v = fmaxf(v, 0.f);
            out[(size_t)row * FDIM + col] = v;
        }
    }
}

// GCN fused: xw = A @ W; xw_out <- xw; acc_out <- xw*dinv[row]^2 + b3[col]
__global__ void gemm_gcn_kernel(const float* __restrict__ A,
                                const float* __restrict__ W,
                                const float* __restrict__ dinv,
                                const float* __restrict__ b3,
                                float* __restrict__ xw_out,
                                float* __restrict__ acc_out, int nrows) {
    __shared__ float lw[FDIM * FDIM];
    __shared__ float lb[FDIM];
    int t = threadIdx.x;
    stage_to_lds(lw, W, FDIM * FDIM, t);
    stage_to_lds(lb, b3, FDIM, t);
    stage_wait();

    int wave = (blockIdx.x * blockDim.x + t) >> 5;
    int m0 = wave * 16;
    if (m0 + 16 > nrows) return;
    int lane = t & 31;
    int half = lane >> 4;
    int lm = lane & 15;

    v8f acc[4] = {};
    const float* arow = A + (size_t)(m0 + lm) * FDIM;

    for (int kk = 0; kk < FDIM; kk += 4) {
        int ko = kk + half * 2;
        v2f a = *(const v2f*)(arow + ko);
#pragma unroll
        for (int nt = 0; nt < 4; ++nt) {
            int n = nt * 16 + lm;
            v2f b;
            b.x = lw[ko * FDIM + n];
            b.y = lw[(ko + 1) * FDIM + n];
            acc[nt] = __builtin_amdgcn_wmma_f32_16x16x4_f32(
                false, a, false, b, (short)0, acc[nt], false, false);
        }
    }

#pragma unroll
    for (int r = 0; r < 8; ++r) {
        int row = m0 + r + half * 8;
        float di = dinv[row];
        float di2 = di * di;
#pragma unroll
        for (int nt = 0; nt < 4; ++nt) {
            int col = nt * 16 + lm;
            float v = acc[nt][r];
            xw_out[(size_t)row * FDIM + col] = v;
            acc_out[(size_t)row * FDIM + col] = v * di2 + lb[col];
        }
    }
}

// out[N,10] = A[N,64] @ W[64,10] + bias; single 16-col WMMA tile, cols>=10 masked
__global__ void head_gemm_kernel(const float* __restrict__ A,
                                 const float* __restrict__ W,
                                 const float* __restrict__ bias,
                                 float* __restrict__ out, int nrows) {
    __shared__ float lw[FDIM * CDIM];  // 640 floats
    __shared__ float lb[CDIM];
    int t = threadIdx.x;
    stage_to_lds(lw, W, FDIM * CDIM, t);
    if (t < CDIM) lb[t] = bias[t];
    stage_wait();

    int wave = (blockIdx.x * blockDim.x + t) >> 5;
    int m0 = wave * 16;
    if (m0 + 16 > nrows) return;
    int lane = t & 31;
    int half = lane >> 4;
    int lm = lane & 15;

    v8f acc = {};
    const float* arow = A + (size_t)(m0 + lm) * FDIM;

    for (int kk = 0; kk < FDIM; kk += 4) {
        int ko = kk + half * 2;
        v2f a = *(const v2f*)(arow + ko);
        v2f b;
        b.x = (lm < CDIM) ? lw[ko * CDIM + lm] : 0.f;
        b.y = (lm < CDIM) ? lw[(ko + 1) * CDIM + lm] : 0.f;
        acc = __builtin_amdgcn_wmma_f32_16x16x4_f32(
            false, a, false, b, (short)0, acc, false, false);
    }

    if (lm < CDIM) {
        float bb = lb[lm];
#pragma unroll
        for (int r = 0; r < 8; ++r) {
            int row = m0 + r + half * 8;
            out[(size_t)row * CDIM + lm] = acc[r] + bb;
        }
    }
}

// ------------------------------- launcher ----------------------------------

extern "C" void kernel_launch(void* const* d_in, const int* in_sizes, int n_in,
                              void* d_out, int out_size, void* d_ws, size_t ws_size,
                              hipStream_t stream) {
    const float*     x       = (const float*)d_in[0];
    const long long* eidx    = (const long long*)d_in[1];  // [2,E] int64
    const float*     W1_rel  = (const float*)d_in[2];
    const float*     b1      = (const float*)d_in[3];
    const float*     W1_root = (const float*)d_in[4];
    const float*     W2_rel  = (const float*)d_in[5];
    const float*     b2      = (const float*)d_in[6];
    const float*     W2_root = (const float*)d_in[7];
    const float*     W3      = (const float*)d_in[8];
    const float*     b3      = (const float*)d_in[9];
    const float*     Wlin    = (const float*)d_in[10];
    const float*     blin    = (const float*)d_in[11];
    float*           out     = (float*)d_out;

    // workspace layout (floats): bufA,bufB,bufC = N*64 each; bufD = N
    float* bufA = (float*)d_ws;                     // agg1 / agg2 / gcn-acc
    float* bufB = bufA + (size_t)NN * FDIM;         // h1, then xw
    float* bufC = bufB + (size_t)NN * FDIM;         // h2
    float* bufD = bufC + (size_t)NN * FDIM;         // deg -> dinv

    const int nElem = NN * FDIM;                    // 6.4M
    const int zeroBlocks    = (nElem / 4 + BLK - 1) / BLK;
    const int scatterBlocks = (EE * 32 + BLK - 1) / BLK;
    const int tiles = NN / 16;                      // 6250 (N divisible by 16)
    const int gemmBlocks = (tiles * 32 + BLK - 1) / BLK;
    const int nodeBlocks = (NN + BLK - 1) / BLK;
    const int edgeBlocks = (EE + BLK - 1) / BLK;

    // ---- Layer 1: GraphConv(x) -> h1 (bufB) ----
    zero_f4_kernel<<<zeroBlocks, BLK, 0, stream>>>((float4*)bufA, nElem / 4);
    scatter_feat_kernel<<<scatterBlocks, BLK, 0, stream>>>(x, eidx, bufA, EE);
    gemm_dual_kernel<<<gemmBlocks, BLK, 0, stream>>>(bufA, W1_rel, x, W1_root,
                                                     b1, bufB, NN, 1);

    // ---- Layer 2: GraphConv(h1) -> h2 (bufC) ----
    zero_f4_kernel<<<zeroBlocks, BLK, 0, stream>>>((float4*)bufA, nElem / 4);
    scatter_feat_kernel<<<scatterBlocks, BLK, 0, stream>>>(bufB, eidx, bufA, EE);
    gemm_dual_kernel<<<gemmBlocks, BLK, 0, stream>>>(bufA, W2_rel, bufB, W2_root,
                                                     b2, bufC, NN, 1);

    // ---- Layer 3: GCNConv(h2) ----
    fill1_kernel<<<nodeBlocks, BLK, 0, stream>>>(bufD, NN);
    scatter_deg_kernel<<<edgeBlocks, BLK, 0, stream>>>(eidx, bufD, EE);
    rsqrt_kernel<<<nodeBlocks, BLK, 0, stream>>>(bufD, NN);
    // xw = h2 @ W3 -> bufB; acc (self-loop + bias) -> bufA (fused epilogue)
    gemm_gcn_kernel<<<gemmBlocks, BLK, 0, stream>>>(bufC, W3, bufD, b3,
                                                    bufB, bufA, NN);
    scatter_norm_kernel<<<scatterBlocks, BLK, 0, stream>>>(bufB, eidx, bufD,
                                                           bufA, EE);

    // ---- Head: out = acc @ Wlin + blin ----
    head_gemm_kernel<<<gemmBlocks, BLK, 0, stream>>>(bufA, Wlin, blin, out, NN);
}